// GATLayerWithWeights_53077205844651
// MI455X (gfx1250) — compile-verified
//
#include <hip/hip_runtime.h>
#include <math.h>

#define NN 10000
#define NE 320000
#define DF 256
#define MT (NN / 16)   // 625 row tiles (exact)
#define NT (DF / 16)   // 16 col tiles

typedef __attribute__((ext_vector_type(2))) float v2f;
typedef __attribute__((ext_vector_type(8))) float v8f;

// ---------------------------------------------------------------- init
__global__ __launch_bounds__(256) void k_init(float* __restrict__ out_all, long long n_out,
                                              float* __restrict__ seg_max,
                                              float* __restrict__ denom,
                                              float* __restrict__ deg) {
  long long i = (long long)blockIdx.x * blockDim.x + threadIdx.x;
  long long stride = (long long)gridDim.x * blockDim.x;
  for (long long j = i; j < n_out; j += stride) out_all[j] = 0.0f;
  for (long long j = i; j < NN; j += stride) {
    seg_max[j] = -INFINITY;
    denom[j] = 0.0f;
    deg[j] = 0.0f;
  }
}

// ---------------------------------------------------------------- GEMM: h = x @ W^T (f32 WMMA)
// One wave32 per 16x16 tile of h. A = 16x4 x-tile, B = 4x16 W^T-tile, 64 K-steps.
__global__ __launch_bounds__(256) void k_gemm_wmma(const float* __restrict__ x,
                                                   const float* __restrict__ Wm,
                                                   float* __restrict__ h) {
  const int lane = threadIdx.x & 31;
  const int wave = (int)((blockIdx.x * blockDim.x + threadIdx.x) >> 5);
  const int tm = wave >> 4;   // row tile 0..624
  const int tn = wave & 15;   // col tile 0..15
  if (tm >= MT) return;
  const int r0 = tm * 16, c0 = tn * 16;
  const int half = lane >> 4;  // 0: K pair {0,1}; 1: K pair {2,3}
  const int l = lane & 15;

  v8f acc = {};
#if __has_builtin(__builtin_amdgcn_wmma_f32_16x16x4_f32)
  const v2f* xv = reinterpret_cast<const v2f*>(x);
  const v2f* wv = reinterpret_cast<const v2f*>(Wm);
  const int arow = (r0 + l) * (DF / 2);  // x row, in v2f units
  const int brow = (c0 + l) * (DF / 2);  // W row (== column of B), in v2f units
#pragma unroll 4
  for (int k0 = 0; k0 < DF; k0 += 4) {
    v2f a = xv[arow + (k0 >> 1) + half];   // x[r0+l][k0+2*half .. +1]
    v2f b = wv[brow + (k0 >> 1) + half];   // W[c0+l][k0+2*half .. +1] == B[k][c0+l]
    acc = __builtin_amdgcn_wmma_f32_16x16x4_f32(false, a, false, b, (short)0, acc,
                                                false, false);
  }
#else
  // scalar fallback matching the D-matrix lane layout
  {
    const int col = c0 + l;
#pragma unroll
    for (int r = 0; r < 8; ++r) {
      const int row = r0 + half * 8 + r;
      float s = 0.0f;
      for (int k = 0; k < DF; ++k) s = fmaf(x[row * DF + k], Wm[col * DF + k], s);
      acc[r] = s;
    }
  }
#endif
  // D layout: VGPR r -> M = r (lanes 0-15) / M = r+8 (lanes 16-31), N = l
  float* hp = h + (size_t)(r0 + half * 8) * DF + c0 + l;
#pragma unroll
  for (int r = 0; r < 8; ++r) hp[(size_t)r * DF] = acc[r];
}

// ---------------------------------------------------------------- per-node attention dots
__global__ __launch_bounds__(256) void k_node_dots(const float* __restrict__ h,
                                                   const float* __restrict__ att_src,
                                                   const float* __restrict__ att_dst,
                                                   float* __restrict__ s_src,
                                                   float* __restrict__ s_dst) {
  const int node = (int)((blockIdx.x * blockDim.x + threadIdx.x) >> 5);
  const int lane = threadIdx.x & 31;
  if (node >= NN) return;
  float ss = 0.0f, sd = 0.0f;
  const float* hp = h + (size_t)node * DF;
#pragma unroll
  for (int f = lane; f < DF; f += 32) {
    float hv = hp[f];
    ss = fmaf(hv, att_src[f], ss);
    sd = fmaf(hv, att_dst[f], sd);
  }
#pragma unroll
  for (int off = 16; off > 0; off >>= 1) {
    ss += __shfl_xor(ss, off, 32);
    sd += __shfl_xor(sd, off, 32);
  }
  if (lane == 0) {
    s_src[node] = ss;
    s_dst[node] = sd;
  }
}

// ---------------------------------------------------------------- float atomic max (sign-split trick)
__device__ inline void atomicMaxF(float* addr, float v) {
  if (v >= 0.0f)
    atomicMax((int*)addr, __float_as_int(v));
  else
    atomicMin((unsigned int*)addr, (unsigned int)__float_as_int(v));
}

// ---------------------------------------------------------------- per-edge logits + segment max + degree
__global__ __launch_bounds__(256) void k_edge_score(const float* __restrict__ s_src,
                                                    const float* __restrict__ s_dst,
                                                    const int* __restrict__ src,
                                                    const int* __restrict__ dst,
                                                    float* __restrict__ escore,
                                                    float* __restrict__ seg_max,
                                                    float* __restrict__ deg) {
  const int e = blockIdx.x * blockDim.x + threadIdx.x;
  if (e >= NE) return;
  const int s = src[e], d = dst[e];
  float sc = s_src[s] + s_dst[d];
  sc = (sc >= 0.0f) ? sc : 0.2f * sc;  // LeakyReLU(0.2)
  escore[e] = sc;
  atomicMaxF(&seg_max[d], sc);
  atomicAdd(&deg[d], 1.0f);
}

// ---------------------------------------------------------------- per-edge exp + segment sum
__global__ __launch_bounds__(256) void k_edge_exp(float* __restrict__ escore,
                                                  const float* __restrict__ seg_max,
                                                  const int* __restrict__ dst,
                                                  float* __restrict__ denom) {
  const int e = blockIdx.x * blockDim.x + threadIdx.x;
  if (e >= NE) return;
  const int d = dst[e];
  const float ex = __expf(escore[e] - seg_max[d]);
  escore[e] = ex;
  atomicAdd(&denom[d], ex);
}

// ---------------------------------------------------------------- aggregation + dense attention matrix
__global__ __launch_bounds__(256) void k_aggregate(const float* __restrict__ h,
                                                   const float* __restrict__ escore,
                                                   const float* __restrict__ denom,
                                                   const int* __restrict__ src,
                                                   const int* __restrict__ dst,
                                                   float* __restrict__ out,
                                                   float* __restrict__ attmat) {
  const int e = (int)((blockIdx.x * blockDim.x + threadIdx.x) >> 5);
  const int lane = threadIdx.x & 31;
  if (e >= NE) return;
  const int s = src[e], d = dst[e];
  const float alpha = escore[e] / denom[d];
  if (lane == 0) attmat[(long long)s * NN + d] = alpha;
  const float* hp = h + (size_t)s * DF;
  float* op = out + (size_t)d * DF;
#pragma unroll
  for (int f = lane; f < DF; f += 32) atomicAdd(&op[f], alpha * hp[f]);
}

// ---------------------------------------------------------------- isolated nodes keep h
__global__ __launch_bounds__(256) void k_isolated(const float* __restrict__ h,
                                                  const float* __restrict__ deg,
                                                  float* __restrict__ out) {
  const int node = (int)((blockIdx.x * blockDim.x + threadIdx.x) >> 5);
  const int lane = threadIdx.x & 31;
  if (node >= NN) return;
  if (deg[node] != 0.0f) return;
  const float* hp = h + (size_t)node * DF;
  float* op = out + (size_t)node * DF;
#pragma unroll
  for (int f = lane; f < DF; f += 32) op[f] = hp[f];
}

// ================================================================ launch
extern "C" void kernel_launch(void* const* d_in, const int* in_sizes, int n_in,
                              void* d_out, int out_size, void* d_ws, size_t ws_size,
                              hipStream_t stream) {
  const float* x = (const float*)d_in[0];
  const float* Wm = (const float*)d_in[1];
  const float* att_src = (const float*)d_in[2];
  const float* att_dst = (const float*)d_in[3];
  const int* ei = (const int*)d_in[4];
  const int* src = ei;
  const int* dst = ei + NE;

  float* out = (float*)d_out;                  // [NN*DF]
  float* attmat = out + (size_t)NN * DF;       // [NN*NN]

  float* ws = (float*)d_ws;
  float* h = ws;        ws += (size_t)NN * DF;  // 2.56M
  float* s_src = ws;    ws += NN;
  float* s_dst = ws;    ws += NN;
  float* seg_max = ws;  ws += NN;
  float* denom = ws;    ws += NN;
  float* deg = ws;      ws += NN;
  float* escore = ws;   ws += NE;

  // 1. zero out + attention matrix, init reductions
  k_init<<<4096, 256, 0, stream>>>(out, (long long)out_size, seg_max, denom, deg);

  // 2. h = x @ W^T : 10000 tile-waves, 8 waves per 256-thread block
  k_gemm_wmma<<<(MT * NT) / 8, 256, 0, stream>>>(x, Wm, h);

  // 3. per-node attention dot products (wave per node)
  k_node_dots<<<(NN * 32 + 255) / 256, 256, 0, stream>>>(h, att_src, att_dst, s_src, s_dst);

  // 4. per-edge logits + segment max + degree
  k_edge_score<<<(NE + 255) / 256, 256, 0, stream>>>(s_src, s_dst, src, dst, escore, seg_max, deg);

  // 5. per-edge exp + segment sum
  k_edge_exp<<<(NE + 255) / 256, 256, 0, stream>>>(escore, seg_max, dst, denom);

  // 6. weighted aggregation + dense attention matrix (wave per edge)
  k_aggregate<<<(NE * 32 + 255) / 256, 256, 0, stream>>>(h, escore, denom, src, dst, out, attmat);

  // 7. isolated nodes copy h
  k_isolated<<<(NN * 32 + 255) / 256, 256, 0, stream>>>(h, deg, out);
}